// DecoderTP_accu_53257594471032
// MI455X (gfx1250) — compile-verified
//
#include <hip/hip_runtime.h>

// ---------------- problem constants (match reference) ----------------
static constexpr int kE        = 256;
static constexpr int kS        = 16;
static constexpr int kN        = 8192;
static constexpr int kNSrc     = 10000;
static constexpr int kMinDst   = 10000;
static constexpr float kTdMax  = 5000.0f;

typedef __attribute__((ext_vector_type(2))) float v2f;
typedef __attribute__((ext_vector_type(8))) float v8f;

// psi * (log1p(exp(-g_psi)) + g_psi), g = gdot + b + alpha*exp(-w_t*td/5000)
__device__ __forceinline__ float hawkes_f(float gdot, float td,
                                          float b, float psi, float alpha, float wt) {
    float g  = gdot + b + alpha * expf(-wt * (td * (1.0f / kTdMax)));
    float gp = g / (psi + 1e-7f);
    gp = fminf(fmaxf(gp, -75.0f), 75.0f);
    return psi * (log1pf(expf(-gp)) + gp);
}

// ---------------- kernel 0: zero accumulators ----------------
__global__ void zero_kernel(float* __restrict__ out, float* __restrict__ ws, int nws) {
    int i = blockIdx.x * blockDim.x + threadIdx.x;
    if (i < nws) ws[i] = 0.0f;
    if (i < 2)   out[i] = 0.0f;
}

// ---------------- kernel 1: survival intensities ----------------
// One wave per (16-column tile, s). 8192 wave-tasks total.
// g[j] = u_non[s*N+j] . Wu + v_non[s*N+j] . Wv   via V_WMMA_F32_16X16X4_F32,
// with W broadcast across all 16 B-columns.
__global__ __launch_bounds__(256) void surv_kernel(
    const float* __restrict__ u_non, const float* __restrict__ v_non,
    const float* __restrict__ W,
    const int*   __restrict__ src,  const int* __restrict__ dst,
    const int*   __restrict__ assoc,
    const float* __restrict__ last_update,
    const float* __restrict__ last_time_pos,
    const float* __restrict__ cur_time,
    const float* __restrict__ td_surv_step,
    const float* __restrict__ b_omega, const float* __restrict__ psi,
    const float* __restrict__ alpha,   const float* __restrict__ w_t,
    float* __restrict__ ws_lam)
{
    __shared__ float ldsW[2 * kE];   // Wu | Wv (512 floats)
    __shared__ float ldsG[8 * 16];   // per-wave dot-product exchange

    for (int i = threadIdx.x; i < 2 * kE; i += blockDim.x) ldsW[i] = W[i];
    __syncthreads();

    const int wave = threadIdx.x >> 5;
    const int lane = threadIdx.x & 31;
    const int task = blockIdx.x * 8 + wave;      // 0 .. N/16*S - 1
    const int s    = task & (kS - 1);
    const int tile = task >> 4;                  // 0 .. 511
    const int j0   = tile * 16;
    const int j    = j0 + (lane & 15);
    const int off2 = (lane >> 4) * 2;            // K sub-offset per A/B layout

    const float* up = u_non + (size_t)(s * kN + j) * kE;
    const float* vp = v_non + (size_t)(s * kN + j) * kE;

    v8f c = {0.f, 0.f, 0.f, 0.f, 0.f, 0.f, 0.f, 0.f};
    #pragma unroll 4
    for (int kk = 0; kk < kE / 4; ++kk) {
        const int k = kk * 4 + off2;
        v2f au = *(const v2f*)(up + k);
        v2f bu = *(const v2f*)(&ldsW[k]);
        c = __builtin_amdgcn_wmma_f32_16x16x4_f32(false, au, false, bu,
                                                  (short)0, c, false, false);
        v2f av = *(const v2f*)(vp + k);
        v2f bv = *(const v2f*)(&ldsW[kE + k]);
        c = __builtin_amdgcn_wmma_f32_16x16x4_f32(false, av, false, bv,
                                                  (short)0, c, false, false);
    }

    // C/D layout: lane L, VGPR p -> D[M = p + 8*(L>=16), N = L&15]; all N identical.
    if ((lane & 15) == 0) {
        const int mb = (lane >> 4) * 8;
        #pragma unroll
        for (int p = 0; p < 8; ++p) ldsG[wave * 16 + mb + p] = c[p];
    }
    float g = ldsG[wave * 16 + (lane & 15)];

    if (lane < 16) {
        const int si = src[j], di = dst[j];
        const float lu_s = last_update[assoc[si]];
        const float lu_d = last_update[assoc[di]];
        const float ltp  = last_time_pos[j];
        const float ct   = cur_time[j];
        const float td_uv = ct - fmaxf(fmaxf(lu_s, lu_d), ltp);
        const float td    = td_surv_step[s * kN + j] * td_uv;
        const float lam   = hawkes_f(g, td, b_omega[0], psi[0], alpha[0], w_t[0]);
        atomicAdd(&ws_lam[j], lam);
    }
}

// ---------------- kernel 2: event intensities + finalize ----------------
__global__ __launch_bounds__(256) void uv_kernel(
    const float* __restrict__ emb,  const int* __restrict__ assoc,
    const int*   __restrict__ src,  const int* __restrict__ dst,
    const float* __restrict__ last_update,
    const float* __restrict__ cur_time,
    const float* __restrict__ last_time_pos,
    const float* __restrict__ event_accu,
    const float* __restrict__ W,
    const float* __restrict__ b_omega, const float* __restrict__ psi,
    const float* __restrict__ alpha,   const float* __restrict__ w_t,
    const float* __restrict__ ws_lam,  float* __restrict__ out)
{
    __shared__ float ldsW[2 * kE];
    __shared__ float ldsG[8 * 16];

    for (int i = threadIdx.x; i < 2 * kE; i += blockDim.x) ldsW[i] = W[i];
    __syncthreads();

    const int wave = threadIdx.x >> 5;
    const int lane = threadIdx.x & 31;
    const int tile = blockIdx.x * 8 + wave;      // 0 .. 511
    const int j0   = tile * 16;
    const int j    = j0 + (lane & 15);
    const int off2 = (lane >> 4) * 2;

    const int si = src[j], di = dst[j];
    const int as = assoc[si], ad = assoc[di];
    const float* up = emb + (size_t)as * kE;
    const float* vp = emb + (size_t)ad * kE;

    v8f c = {0.f, 0.f, 0.f, 0.f, 0.f, 0.f, 0.f, 0.f};
    #pragma unroll 4
    for (int kk = 0; kk < kE / 4; ++kk) {
        const int k = kk * 4 + off2;
        v2f au = *(const v2f*)(up + k);
        v2f bu = *(const v2f*)(&ldsW[k]);
        c = __builtin_amdgcn_wmma_f32_16x16x4_f32(false, au, false, bu,
                                                  (short)0, c, false, false);
        v2f av = *(const v2f*)(vp + k);
        v2f bv = *(const v2f*)(&ldsW[kE + k]);
        c = __builtin_amdgcn_wmma_f32_16x16x4_f32(false, av, false, bv,
                                                  (short)0, c, false, false);
    }

    if ((lane & 15) == 0) {
        const int mb = (lane >> 4) * 8;
        #pragma unroll
        for (int p = 0; p < 8; ++p) ldsG[wave * 16 + mb + p] = c[p];
    }
    float g = ldsG[wave * 16 + (lane & 15)];

    float vl = 0.0f, vs = 0.0f;
    if (lane < 16) {
        const float lu_s = last_update[as];
        const float lu_d = last_update[ad];
        const float lmax = fmaxf(lu_s, lu_d);
        const float ltp  = last_time_pos[j];
        const float use_accu = (ltp >= lmax) ? 1.0f : 0.0f;
        const float td_uv = cur_time[j] - fmaxf(lmax, ltp);
        const float lam_uv = hawkes_f(g, td_uv, b_omega[0], psi[0], alpha[0], w_t[0]);
        vl = -logf(lam_uv + 1e-7f);
        const float mean = ws_lam[j] * (1.0f / kS);
        const float acc  = event_accu[(size_t)si * kNSrc + (di - kMinDst)];
        vs = mean * td_uv + use_accu * acc;
    }

    // full 32-lane tree reduction (lanes >=16 carry zeros)
    #pragma unroll
    for (int off = 16; off > 0; off >>= 1) {
        vl += __shfl_down(vl, off, 32);
        vs += __shfl_down(vs, off, 32);
    }
    if (lane == 0) {
        atomicAdd(&out[0], vl * (1.0f / kN));   // loss_lambda / n
        atomicAdd(&out[1], vs * (1.0f / kN));   // loss_surv / n
    }
}

// ---------------- host launcher ----------------
extern "C" void kernel_launch(void* const* d_in, const int* in_sizes, int n_in,
                              void* d_out, int out_size, void* d_ws, size_t ws_size,
                              hipStream_t stream) {
    const float* all_emb       = (const float*)d_in[0];
    const int*   assoc         = (const int*)  d_in[1];
    const int*   src           = (const int*)  d_in[2];
    const int*   pos_dst       = (const int*)  d_in[3];
    const float* last_update   = (const float*)d_in[4];
    const float* cur_time      = (const float*)d_in[5];
    const float* u_non         = (const float*)d_in[6];
    const float* v_non         = (const float*)d_in[7];
    const float* last_time_pos = (const float*)d_in[8];
    const float* td_surv_step  = (const float*)d_in[9];
    const float* event_accu    = (const float*)d_in[10];
    const float* W_omega       = (const float*)d_in[11];
    const float* b_omega       = (const float*)d_in[12];
    const float* psi           = (const float*)d_in[13];
    const float* alpha         = (const float*)d_in[14];
    const float* w_t           = (const float*)d_in[15];
    float* out = (float*)d_out;
    float* ws  = (float*)d_ws;   // kN floats of per-column lambda sums

    zero_kernel<<<(kN + 255) / 256, 256, 0, stream>>>(out, ws, kN);

    // (N/16)*S = 8192 wave-tasks, 8 waves per 256-thread block -> 1024 blocks
    surv_kernel<<<(kN / 16) * kS / 8, 256, 0, stream>>>(
        u_non, v_non, W_omega, src, pos_dst, assoc, last_update,
        last_time_pos, cur_time, td_surv_step,
        b_omega, psi, alpha, w_t, ws);

    // N/16 = 512 wave-tiles -> 64 blocks
    uv_kernel<<<(kN / 16) / 8, 256, 0, stream>>>(
        all_emb, assoc, src, pos_dst, last_update, cur_time, last_time_pos,
        event_accu, W_omega, b_omega, psi, alpha, w_t, ws, out);
}